// GCE_GNN_26104811225296
// MI455X (gfx1250) — compile-verified
//
#include <hip/hip_runtime.h>
#include <hip/hip_bf16.h>
#include <math.h>

#define B_ 128
#define L_ 48
#define K_ 12
#define V_ 500000
#define D_ 100
#define N_ (B_*L_)

typedef __attribute__((ext_vector_type(2))) float v2f;
typedef __attribute__((ext_vector_type(8))) float v8f;
typedef __attribute__((ext_vector_type(4))) unsigned int v4u;
typedef __attribute__((ext_vector_type(8))) int v8i;
typedef __attribute__((ext_vector_type(4))) int v4i;

// ---------------- workspace layout (bytes) ----------------
// mask      : B*L*L bytes          @ 0         (294912)
// s_mean    : B floats             @ 294912    (512)
// hidden    : N*D floats           @ 295424    (2457600)
// h_global  : N*D floats           @ 2753024   (2457600)
// h         : N*D floats           @ 5210624   (2457600)
// hmean     : B*D floats           @ 7668224   (51200)
// S         : B*D floats           @ 7719424   (51200)

__global__ void k_zero_mask(unsigned int* m, int nwords) {
  int i = blockIdx.x * blockDim.x + threadIdx.x;
  if (i < nwords) m[i] = 0u;
}

__global__ void k_build_mask(const int* __restrict__ ei, int E, unsigned char* __restrict__ mask) {
  int e = blockIdx.x * blockDim.x + threadIdx.x;
  if (e >= E) return;
  int s = ei[e], d = ei[E + e];
  int bs = s / L_, bd = d / L_;
  if (bs != bd) return;                 // edges are within-session by construction
  int li = s - bs * L_, lj = d - bd * L_;
  mask[bs * L_ * L_ + li * L_ + lj] = 1;
  mask[bs * L_ * L_ + lj * L_ + li] = 1;  // undirected (adj | adj.T)
}

__global__ void k_gather_hidden(const int* __restrict__ x, const float* __restrict__ emb,
                                float* __restrict__ hidden) {
  int idx = blockIdx.x * blockDim.x + threadIdx.x;
  if (idx >= N_ * D_) return;
  int i = idx / D_, d = idx - i * D_;
  hidden[idx] = emb[(size_t)x[i] * D_ + d];
}

__global__ void k_smean(const int* __restrict__ x, float* __restrict__ s_mean) {
  __shared__ float red[64];
  int b = blockIdx.x, t = threadIdx.x;
  red[t] = (t < L_) ? (float)x[b * L_ + t] : 0.f;
  __syncthreads();
  for (int s = 32; s > 0; s >>= 1) {
    if (t < s) red[t] += red[t + s];
    __syncthreads();
  }
  if (t == 0) s_mean[b] = red[0] / (float)L_;
}

// one 128-thread workgroup per node: K=12 neighbor attention + h_global MLP
__global__ __launch_bounds__(128) void k_global(
    const int* __restrict__ nbr_ids, const float* __restrict__ nbr_w,
    const float* __restrict__ emb,
    const float* __restrict__ w1_W, const float* __restrict__ w1_b, const float* __restrict__ q1_w,
    const float* __restrict__ w2_W, const float* __restrict__ w2_b,
    const float* __restrict__ s_mean, const float* __restrict__ hidden,
    float* __restrict__ h_global) {
  __shared__ float sw1[101 * 101];      // w1_W staged: reused 12x per node
  __shared__ float snemb[K_ * D_];
  __shared__ float shid[D_];
  __shared__ float shn[D_];
  __shared__ float swsoft[K_];
  __shared__ float sa[K_];
  __shared__ float salpha[K_];
  int node = blockIdx.x, tid = threadIdx.x;
  int b = node / L_;

  for (int i = tid; i < 101 * 101; i += 128) sw1[i] = w1_W[i];
  for (int i = tid; i < K_ * D_; i += 128) {
    int k = i / D_, g = i - k * D_;
    snemb[i] = emb[(size_t)nbr_ids[node * K_ + k] * D_ + g];
  }
  for (int i = tid; i < D_; i += 128) shid[i] = hidden[node * D_ + i];
  if (tid < K_) sa[tid] = 0.f;
  if (tid == 0) {
    float w[K_], mx = -1e30f;
    for (int k = 0; k < K_; ++k) { w[k] = nbr_w[node * K_ + k]; mx = fmaxf(mx, w[k]); }
    float s = 0.f;
    for (int k = 0; k < K_; ++k) { w[k] = __expf(w[k] - mx); s += w[k]; }
    for (int k = 0; k < K_; ++k) swsoft[k] = w[k] / s;
  }
  __syncthreads();

  float sm = s_mean[b];
  if (tid < 101) {
    int f = tid;
    float qf = q1_w[f], bf = w1_b[f];
    for (int k = 0; k < K_; ++k) {
      float acc = bf;
      for (int g = 0; g < D_; ++g) acc += sm * snemb[k * D_ + g] * sw1[g * 101 + f];
      acc += swsoft[k] * sw1[100 * 101 + f];
      float t = acc > 0.f ? acc : 0.01f * acc;   // leaky_relu(0.01)
      atomicAdd(&sa[k], t * qf);
    }
  }
  __syncthreads();
  if (tid == 0) {
    float mx = -1e30f;
    for (int k = 0; k < K_; ++k) mx = fmaxf(mx, sa[k]);
    float s = 0.f, e[K_];
    for (int k = 0; k < K_; ++k) { e[k] = __expf(sa[k] - mx); s += e[k]; }
    for (int k = 0; k < K_; ++k) salpha[k] = e[k] / s;
  }
  __syncthreads();
  if (tid < D_) {
    float acc = 0.f;
    for (int k = 0; k < K_; ++k) acc += salpha[k] * snemb[k * D_ + tid];
    shn[tid] = acc;
  }
  __syncthreads();
  if (tid < D_) {
    int d = tid;
    float acc = w2_b[d];
    for (int g = 0; g < D_; ++g) acc += shid[g] * w2_W[g * D_ + d];
    for (int g = 0; g < D_; ++g) acc += shn[g] * w2_W[(D_ + g) * D_ + d];
    h_global[node * D_ + d] = fmaxf(acc, 0.f);
  }
}

// one 256-thread workgroup per session: 48x48 masked attention (block-diagonal)
__global__ __launch_bounds__(256) void k_local(
    const float* __restrict__ a_w, const unsigned char* __restrict__ mask,
    const float* __restrict__ hidden, const float* __restrict__ h_global,
    float* __restrict__ h, float* __restrict__ hmean) {
  __shared__ float sh[L_ * D_];         // 19.2 KB
  __shared__ float ssc[L_ * L_];        // 9.2 KB
  __shared__ float shmean[D_];
  int b = blockIdx.x, tid = threadIdx.x;
  for (int i = tid; i < L_ * D_; i += 256) sh[i] = hidden[b * L_ * D_ + i];
  if (tid < D_) shmean[tid] = 0.f;
  __syncthreads();
  for (int p = tid; p < L_ * L_; p += 256) {
    int i = p / L_, j = p - i * L_;
    float acc = 0.f;
    for (int d = 0; d < D_; ++d) acc += sh[i * D_ + d] * a_w[d] * sh[j * D_ + d];
    float sc = acc > 0.f ? acc : 0.2f * acc;     // leaky_relu(0.2)
    ssc[p] = mask[b * L_ * L_ + p] ? sc : -1e9f;
  }
  __syncthreads();
  if (tid < L_) {                                // row softmax
    float mx = -1e30f;
    for (int j = 0; j < L_; ++j) mx = fmaxf(mx, ssc[tid * L_ + j]);
    float s = 0.f;
    for (int j = 0; j < L_; ++j) { float e = __expf(ssc[tid * L_ + j] - mx); ssc[tid * L_ + j] = e; s += e; }
    float inv = 1.f / s;
    for (int j = 0; j < L_; ++j) ssc[tid * L_ + j] *= inv;
  }
  __syncthreads();
  for (int q = tid; q < L_ * D_; q += 256) {
    int i = q / D_, d = q - i * D_;
    float acc = 0.f;
    for (int j = 0; j < L_; ++j) acc += ssc[i * L_ + j] * sh[j * D_ + d];
    float hv = acc + h_global[b * L_ * D_ + q];
    h[b * L_ * D_ + q] = hv;
    atomicAdd(&shmean[d], hv);
  }
  __syncthreads();
  if (tid < D_) hmean[b * D_ + tid] = shmean[tid] / (float)L_;
}

// one 128-thread workgroup per session: z/beta/S with session-constant projections hoisted
__global__ __launch_bounds__(128) void k_scores(
    const float* __restrict__ h, const float* __restrict__ hmean,
    const float* __restrict__ pos_w,
    const float* __restrict__ w3_W, const float* __restrict__ w3_b,
    const float* __restrict__ q2_w,
    const float* __restrict__ w4_W, const float* __restrict__ w5_W, const float* __restrict__ w5_b,
    float* __restrict__ S) {
  __shared__ float shh[L_ * D_];
  __shared__ float sz[L_ * D_];
  __shared__ float sposproj[D_];
  __shared__ float ssproj[D_];
  __shared__ float shm[D_];
  __shared__ float sbeta[L_];
  int b = blockIdx.x, tid = threadIdx.x;
  for (int i = tid; i < L_ * D_; i += 128) shh[i] = h[b * L_ * D_ + i];
  if (tid < D_) shm[tid] = hmean[b * D_ + tid];
  if (tid < L_) sbeta[tid] = 0.f;
  __syncthreads();
  if (tid < D_) {
    int d = tid;
    float pp = w3_b[d], sp = w5_b[d];
    for (int g = 0; g < D_; ++g) pp += pos_w[b * D_ + g] * w3_W[(D_ + g) * D_ + d];
    for (int g = 0; g < D_; ++g) sp += shm[g] * w5_W[g * D_ + d];
    sposproj[d] = pp;
    ssproj[d] = sp;
  }
  __syncthreads();
  for (int q = tid; q < L_ * D_; q += 128) {
    int i = q / D_, d = q - i * D_;
    float acc = sposproj[d];
    for (int g = 0; g < D_; ++g) acc += shh[i * D_ + g] * w3_W[g * D_ + d];
    sz[q] = tanhf(acc);
  }
  __syncthreads();
  for (int q = tid; q < L_ * D_; q += 128) {
    int i = q / D_, d = q - i * D_;
    float acc = ssproj[d];
    for (int g = 0; g < D_; ++g) acc += sz[i * D_ + g] * w4_W[g * D_ + d];
    float u = 1.f / (1.f + __expf(-acc));
    atomicAdd(&sbeta[i], u * q2_w[d]);
  }
  __syncthreads();
  if (tid < D_) {
    float acc = 0.f;
    for (int i = 0; i < L_; ++i) acc += sbeta[i] * shh[i * D_ + tid];
    S[b * D_ + tid] = acc;
  }
}

// S[128,100] @ emb^T -> out[128, 500000] via V_WMMA_F32_16X16X4_F32.
// Each 256-thread block (8 waves) handles 128 columns. The emb tile
// (128 rows x 100 f32, row-stride 100) is DMA'd into LDS by the Tensor Data
// Mover: one descriptor issued by wave 0, tracked with TENSORcnt. Tail blocks
// rely on the D#'s OOB-read-returns-zero semantics (stores stay guarded).
__global__ __launch_bounds__(256) void k_out(const float* __restrict__ S,
                                             const float* __restrict__ emb,
                                             float* __restrict__ out) {
  __shared__ float sE[128 * D_];        // 51.2 KB
  int tid = threadIdx.x;
  int colBase = blockIdx.x * 128;
  if (tid == 0) __builtin_prefetch(S, 0, 0);   // global_prefetch_b8 for the A matrix

#if __has_builtin(__builtin_amdgcn_tensor_load_to_lds)
  if (tid < 32) {                       // wave 0 issues the TDM descriptor
    unsigned long long gaddr =
        (unsigned long long)(const void*)(emb + (size_t)colBase * D_);
    unsigned ldsOff =
        (unsigned)(unsigned long long)(__attribute__((address_space(3))) float*)sE;
    unsigned rows = (unsigned)(V_ - colBase);   // tensor rows remaining from tile start
    // ---- D# group 0 (ISA 8.3): count=1, lds_addr, global_addr, type=2 ----
    v4u g0 = { 1u,                                  // [1:0] count=1, no gather
               ldsOff,                              // [63:32] lds_addr (bytes)
               (unsigned)gaddr,                     // [95:64] global_addr lo
               ((unsigned)(gaddr >> 32) & 0x01FFFFFFu) | (2u << 30) };  // addr hi + type=2
    // ---- D# group 1 (ISA 8.4): data_size=4B, dims/strides, 100x128 tile ----
    v8i g1 = { (int)(2u << 16),                     // [17:16] data_size = 4 bytes
               (int)((100u & 0xFFFFu) << 16),       // [63:48] tensor_dim0 lo16 = 100
               (int)((rows & 0xFFFFu) << 16),       // [79:64] dim0 hi=0, [31:16] tensor_dim1 lo16
               (int)((rows >> 16) | (100u << 16)),  // [15:0] tensor_dim1 hi, [31:16] tile_dim0=100
               (int)128,                            // [15:0] tile_dim1=128, tile_dim2=0
               (int)100,                            // tensor_dim0_stride lo32 = 100
               0, 0 };                              // stride hi / tensor_dim1_stride (unused, 2D)
    v4i gz4 = { 0, 0, 0, 0 };                       // groups 2/3 unused (<=2D tensor)
    v8i gz8 = { 0, 0, 0, 0, 0, 0, 0, 0 };          // extra group (clang-23 6-arg form)
    __builtin_amdgcn_tensor_load_to_lds(g0, g1, gz4, gz4, gz8, 0);
    __builtin_amdgcn_s_wait_tensorcnt(0);
  }
  __syncthreads();                       // publish DMA'd tile to all 8 waves
#else
  for (int i = tid; i < 128 * D_; i += 256) {
    int r = i / D_, c = i - r * D_;
    int v = colBase + r;
    if (v >= V_) v = V_ - 1;
    sE[i] = emb[(size_t)v * D_ + c];
  }
  __syncthreads();
#endif

  int wave = tid >> 5;                  // wave32
  int lane = tid & 31;
  int lh = lane >> 4;
  int l16 = lane & 15;
  int col = wave * 16 + l16;
  v8f acc[8] = {};
  for (int k = 0; k < D_; k += 4) {     // 25 K-steps, exact (100 % 4 == 0)
    v2f bf = *(const v2f*)&sE[col * D_ + k + 2 * lh];
#pragma unroll
    for (int m = 0; m < 8; ++m) {
      v2f af = *(const v2f*)&S[(m * 16 + l16) * D_ + k + 2 * lh];
      acc[m] = __builtin_amdgcn_wmma_f32_16x16x4_f32(
          false, af, false, bf, (short)0, acc[m], false, false);
    }
  }
  int vcol = colBase + wave * 16 + l16;
  if (vcol < V_) {
#pragma unroll
    for (int m = 0; m < 8; ++m) {
#pragma unroll
      for (int r = 0; r < 8; ++r) {
        int row = m * 16 + r + 8 * lh;
        out[(size_t)row * V_ + vcol] = acc[m][r];
      }
    }
  }
}

extern "C" void kernel_launch(void* const* d_in, const int* in_sizes, int n_in,
                              void* d_out, int out_size, void* d_ws, size_t ws_size,
                              hipStream_t stream) {
  const int*   x          = (const int*)d_in[0];
  const int*   edge_index = (const int*)d_in[1];
  const int*   nbr_ids    = (const int*)d_in[2];
  const float* nbr_w      = (const float*)d_in[3];
  const float* emb        = (const float*)d_in[4];
  const float* pos_w      = (const float*)d_in[5];
  const float* a_w        = (const float*)d_in[6];
  const float* w1_W       = (const float*)d_in[7];
  const float* w1_b       = (const float*)d_in[8];
  const float* q1_w       = (const float*)d_in[9];
  const float* w2_W       = (const float*)d_in[10];
  const float* w2_b       = (const float*)d_in[11];
  const float* w3_W       = (const float*)d_in[12];
  const float* w3_b       = (const float*)d_in[13];
  const float* q2_w       = (const float*)d_in[14];
  const float* w4_W       = (const float*)d_in[15];
  const float* w5_W       = (const float*)d_in[16];
  const float* w5_b       = (const float*)d_in[17];
  float* out = (float*)d_out;

  char* ws = (char*)d_ws;
  unsigned char* mask = (unsigned char*)ws;
  float* s_mean  = (float*)(ws + 294912);
  float* hidden  = (float*)(ws + 295424);
  float* h_glob  = (float*)(ws + 2753024);
  float* h       = (float*)(ws + 5210624);
  float* hmean   = (float*)(ws + 7668224);
  float* Smat    = (float*)(ws + 7719424);

  int E = in_sizes[1] / 2;

  k_zero_mask<<<(B_ * L_ * L_ / 4 + 255) / 256, 256, 0, stream>>>((unsigned int*)mask, B_ * L_ * L_ / 4);
  k_build_mask<<<(E + 255) / 256, 256, 0, stream>>>(edge_index, E, mask);
  k_gather_hidden<<<(N_ * D_ + 255) / 256, 256, 0, stream>>>(x, emb, hidden);
  k_smean<<<B_, 64, 0, stream>>>(x, s_mean);
  k_global<<<N_, 128, 0, stream>>>(nbr_ids, nbr_w, emb, w1_W, w1_b, q1_w,
                                   w2_W, w2_b, s_mean, hidden, h_glob);
  k_local<<<B_, 256, 0, stream>>>(a_w, mask, hidden, h_glob, h, hmean);
  k_scores<<<B_, 128, 0, stream>>>(h, hmean, pos_w, w3_W, w3_b, q2_w, w4_W, w5_W, w5_b, Smat);
  k_out<<<(V_ + 127) / 128, 256, 0, stream>>>(Smat, emb, out);
}